// SkipGramNS_90890097918493
// MI455X (gfx1250) — compile-verified
//
#include <hip/hip_runtime.h>
#include <hip/hip_bf16.h>

// SkipGram negative-sampling row-wise dot:
//   out[i] = dot(cxt_weight[cidx[i]], tgt_weight[tidx[i]]), D = 128, fp32.
// Memory-bound (0.25 FLOP/byte). Both embedding tables (102.4 MB) fit in the
// 192 MB L2, so gathers are L2-resident. Strategy: one wave32 computes 16
// pairs using V_WMMA_F32_16X16X4_F32 (exact fp32 FMA, matches reference
// precision), reading each gathered row element exactly once via b128 loads
// and extracting the diagonal of the 16x16 accumulator.

typedef __attribute__((ext_vector_type(2))) float v2f;
typedef __attribute__((ext_vector_type(4))) float v4f;
typedef __attribute__((ext_vector_type(8))) float v8f;

#define D_EMB           128
#define PAIRS_PER_WAVE  16
#define WAVES_PER_BLOCK 8
#define PAIRS_PER_BLOCK (PAIRS_PER_WAVE * WAVES_PER_BLOCK)

__global__ __launch_bounds__(256)
void SkipGramNS_dot_wmma(const int* __restrict__ cidx,
                         const int* __restrict__ tidx,
                         const float* __restrict__ cw,
                         const float* __restrict__ tw,
                         float* __restrict__ out,
                         int n) {
    const int lane = threadIdx.x & 31;   // wave32
    const int wave = threadIdx.x >> 5;   // 0..7
    const int m    = lane & 15;          // row of A (= pair id within tile) / col of B
    const int h    = lane >> 4;          // lane half -> K half of the fragment

    const long long pairBase = (long long)blockIdx.x * PAIRS_PER_BLOCK
                             + (long long)wave * PAIRS_PER_WAVE;
    const long long pair  = pairBase + m;
    const long long pairC = (pair < n) ? pair : 0;   // clamp: keep EXEC full for WMMA

    // Gather row ids (random rows; tables are L2-resident).
    const int crow = cidx[pairC];
    const int trow = tidx[pairC];

    // Column-permuted fragment base: lane half h owns columns {8j+4h .. 8j+4h+3}.
    // Dot products are permutation-invariant over columns, and A/B use the
    // identical per-lane mapping, so the 16x16 diagonal is the exact dot.
    const float* cptr = cw + (size_t)crow * D_EMB + 4 * h;
    const float* tptr = tw + (size_t)trow * D_EMB + 4 * h;

    v8f acc = {};
#pragma unroll
    for (int j = 0; j < 16; ++j) {
        // One b128 load per matrix feeds TWO K=4 WMMA steps.
        v4f a4 = *(const v4f*)(cptr + 8 * j);
        v4f b4 = *(const v4f*)(tptr + 8 * j);

        v2f a0 = {a4.x, a4.y};
        v2f b0 = {b4.x, b4.y};
        acc = __builtin_amdgcn_wmma_f32_16x16x4_f32(
                  false, a0, false, b0, (short)0, acc, false, false);

        v2f a1 = {a4.z, a4.w};
        v2f b1 = {b4.z, b4.w};
        acc = __builtin_amdgcn_wmma_f32_16x16x4_f32(
                  false, a1, false, b1, (short)0, acc, false, false);
    }

    // Diagonal extraction from the 16x16 f32 C/D layout:
    //   (m,m), m<8  -> VGPR m,   lane m        (lanes 0..7)
    //   (m,m), m>=8 -> VGPR m-8, lane m+16     (lanes 24..31)
    const int r  = (lane < 8) ? lane : (lane - 24);        // accumulator slot
    const int om = (lane < 8) ? lane : (lane - 16);        // pair-in-tile id
    float val = acc[0];
#pragma unroll
    for (int i = 1; i < 8; ++i) val = (r == i) ? acc[i] : val;

    if (lane < 8 || lane >= 24) {
        const long long p = pairBase + om;
        if (p < n) out[p] = val;
    }
}

extern "C" void kernel_launch(void* const* d_in, const int* in_sizes, int n_in,
                              void* d_out, int out_size, void* d_ws, size_t ws_size,
                              hipStream_t stream) {
    const int*   cidx = (const int*)  d_in[0];   // context_idxs
    const int*   tidx = (const int*)  d_in[1];   // target_idxs
    const float* cw   = (const float*)d_in[2];   // cxt_weight [V, 128]
    const float* tw   = (const float*)d_in[3];   // tgt_weight [V, 128]
    float*       out  = (float*)      d_out;     // [N, 1]

    const int n = in_sizes[0];
    const int blocks = (n + PAIRS_PER_BLOCK - 1) / PAIRS_PER_BLOCK;
    SkipGramNS_dot_wmma<<<blocks, WAVES_PER_BLOCK * 32, 0, stream>>>(
        cidx, tidx, cw, tw, out, n);
}